// ValueIteration_70970039599283
// MI455X (gfx1250) — compile-verified
//
#include <hip/hip_runtime.h>
#include <math.h>

// ---------------------------------------------------------------------------
// Fused ValueIteration kernel for MI455X (gfx1250, wave32, WMMA), v3.
//
//   dyn(72 x 16px) = Wpad(80x12) @ P(12x16px)   via v_wmma_f32_16x16x4_f32
//   softmax over i (9) per head a (8), q = sum_i p_i * sm_i, out = max_a q
//
// Block = 8 waves = 8x16 pixel tile of one image. Per block we stage:
//   - rv halo tile (10x18, each values/rewards element read ONCE)
//   - zero-padded weight (80x12)  -> branch-free ds_load_b64 A-fragments
// dyn scatter is now UNCONDITIONAL (all 80 padded outputs stored; rows
// 72..79 are exact zeros from the zero-padded weights) -> no exec toggles.
// Grid = 32 images * 20 row-tiles * 10 col-tiles = 6400 blocks.
// ---------------------------------------------------------------------------

typedef __attribute__((ext_vector_type(2))) float v2f;
typedef __attribute__((ext_vector_type(8))) float v8f;

#define B_     32
#define H_     160
#define W_     160
#define HW_    (H_ * W_)
#define A_     8
#define KK_    9          // 3x3 taps
#define NOUT_  72         // A_*KK_
#define WPB_   8          // waves per block = tile rows
#define TCOLS_ 16         // tile cols = pixels per wave
#define MT_    5          // M tiles: 72 -> 80
#define KT_    3          // K tiles: 9 -> 12
#define RV_R_  (WPB_ + 2)     // 10 halo rows
#define RV_C_  (TCOLS_ + 2)   // 18 halo cols
#define RV_CS_ 20             // padded LDS col stride
#define DSTR_  81             // dyn LDS stride: 81 % 64 coprime -> no conflicts

__global__ __launch_bounds__(256)
void ValueIteration_70970039599283_kernel(const float* __restrict__ values,
                                          const float* __restrict__ rewards,
                                          const float* __restrict__ weight,
                                          float* __restrict__ out)
{
    __shared__ __align__(16) float rv_lds[RV_R_][RV_CS_];       // 800 B
    __shared__ __align__(16) float w_lds[MT_ * 16 * (KT_ * 4)]; // 80x12, 3.84 KB
    __shared__ __align__(16) float dyn_lds[WPB_][TCOLS_][DSTR_];// 41.5 KB

    const int tid  = threadIdx.x;
    const int wave = tid >> 5;     // tile row 0..7
    const int lane = tid & 31;
    const int n    = lane & 15;    // pixel (column) within wave
    const int half = lane >> 4;

    // tile decomposition: 10 col-tiles x 20 row-tiles per image, 32 images
    const int bx  = blockIdx.x;
    const int b   = bx / 200;
    const int t   = bx % 200;
    const int ty  = t / 10;
    const int tx  = t % 10;
    const int h0  = ty * WPB_;     // tile origin row
    const int w0  = tx * TCOLS_;   // tile origin col
    const int base = b * HW_;

    // --- stage rv halo tile: 10x18 elements, each global element ONCE -------
    if (tid < RV_R_ * RV_C_) {
        const int er = tid / RV_C_;
        const int ec = tid % RV_C_;
        const int gh = h0 - 1 + er;
        const int gw = w0 - 1 + ec;
        float v = 0.0f;
        if ((unsigned)gh < (unsigned)H_ && (unsigned)gw < (unsigned)W_) {
            const int idx = base + gh * W_ + gw;
            v = values[idx] + rewards[idx];
        }
        rv_lds[er][ec] = v;
    }

    // --- stage zero-padded weight 80x12 (row-major, stride 12) --------------
    for (int idx = tid; idx < MT_ * 16 * (KT_ * 4); idx += 256) {
        const int m = idx / (KT_ * 4);
        const int k = idx % (KT_ * 4);
        w_lds[idx] = (m < NOUT_ && k < KK_) ? weight[m * KK_ + k] : 0.0f;
    }

    __syncthreads();

    // --- B fragments of P(12x16) per 32-bit 16x4 operand layout -------------
    //   lanes 0-15  hold K = 4kt+0 (v0), 4kt+1 (v1)  for pixel = lane
    //   lanes 16-31 hold K = 4kt+2 (v0), 4kt+3 (v1)  for pixel = lane-16
    // Tap k (row-major 3x3): rv_lds[wave + k/3][n + k%3]; K >= 9 -> 0.
    auto tap = [&](int k) -> float { return rv_lds[wave + k / 3][n + k % 3]; };

    v2f bfrag[KT_];
    if (half == 0) {
        bfrag[0] = (v2f){tap(0), tap(1)};
        bfrag[1] = (v2f){tap(4), tap(5)};
        bfrag[2] = (v2f){tap(8), 0.0f};
    } else {
        bfrag[0] = (v2f){tap(2), tap(3)};
        bfrag[1] = (v2f){tap(6), tap(7)};
        bfrag[2] = (v2f){0.0f,   0.0f};
    }

    // --- GEMM: 5 M-tiles x 3 K-tiles of v_wmma_f32_16x16x4_f32 --------------
    // A frag: row m = mt*16 + n (both halves), K = kt*4 + 2*half + {0,1}.
    // Aligned branch-free ds_load_b64 from the padded weight tile.
    v8f acc[MT_];
#pragma unroll
    for (int mt = 0; mt < MT_; ++mt) {
        v8f c = {};
#pragma unroll
        for (int kt = 0; kt < KT_; ++kt) {
            const int m  = mt * 16 + n;
            const int kb = kt * 4 + half * 2;            // even -> 8B aligned
            const v2f af = *(const v2f*)&w_lds[m * (KT_ * 4) + kb];
            // 8 args: (neg_a, A, neg_b, B, c_mod, C, reuse_a, reuse_b)
            c = __builtin_amdgcn_wmma_f32_16x16x4_f32(
                    false, af, false, bfrag[kt], (short)0, c, false, false);
        }
        acc[mt] = c;
    }

    // D layout (32-bit 16x16 C/D): lanes 0-15 VGPR r -> M=r, N=lane;
    // lanes 16-31 VGPR r -> M=r+8, N=lane-16. Unconditional scatter of all
    // 80 padded outputs (o=72..79 are exact zeros, never read back).
#pragma unroll
    for (int mt = 0; mt < MT_; ++mt) {
#pragma unroll
        for (int r = 0; r < 8; ++r) {
            const int o = mt * 16 + r + half * 8;
            dyn_lds[wave][n][o] = acc[mt][r];
        }
    }

    __syncthreads();

    // --- Phase 2: per-pixel softmax + weighted sum + max over heads ---------
    // half 0 -> heads 0-3, half 1 -> heads 4-7; combine via lane shuffle.
    float pv[KK_];
#pragma unroll
    for (int i = 0; i < KK_; ++i) pv[i] = tap(i);

    float best = -3.402823466e38f;
#pragma unroll
    for (int aa = 0; aa < 4; ++aa) {
        const int a = half * 4 + aa;
        float d[KK_];
        float m = -3.402823466e38f;
#pragma unroll
        for (int i = 0; i < KK_; ++i) {
            d[i] = dyn_lds[wave][n][a * KK_ + i];
            m = fmaxf(m, d[i]);
        }
        float s = 0.0f, q = 0.0f;
#pragma unroll
        for (int i = 0; i < KK_; ++i) {
            const float e = __expf(d[i] - m);
            s += e;
            q += pv[i] * e;
        }
        best = fmaxf(best, q / s);
    }

    const float other = __shfl_xor(best, 16, 32);
    best = fmaxf(best, other);

    if (half == 0) out[base + (h0 + wave) * W_ + w0 + n] = best;
}

extern "C" void kernel_launch(void* const* d_in, const int* in_sizes, int n_in,
                              void* d_out, int out_size, void* d_ws, size_t ws_size,
                              hipStream_t stream) {
    (void)in_sizes; (void)n_in; (void)d_ws; (void)ws_size; (void)out_size;
    const float* values  = (const float*)d_in[0];
    const float* rewards = (const float*)d_in[1];
    const float* weight  = (const float*)d_in[2];
    float* out = (float*)d_out;

    const int blocks = B_ * (H_ / WPB_) * (W_ / TCOLS_);  // 6400
    ValueIteration_70970039599283_kernel<<<blocks, 256, 0, stream>>>(
        values, rewards, weight, out);
}